// RoiPooling_29884382445582
// MI455X (gfx1250) — compile-verified
//
#include <hip/hip_runtime.h>

// ROI bilinear pooling (TF1 resize semantics), MI455X / gfx1250.
// Roofline: 120.4 MB write-once output + 37.7 MB image (L2-resident in the
// 192 MB L2; the 482 MB of 4-tap gather reads hit L2, not HBM).
// HBM floor ~158 MB -> ~7 us at 23.3 TB/s. Pure bandwidth; WMMA inapplicable
// (no shared-operand matmul structure in a per-element 4-tap gather).
//
// Design: one block per (roi, py); 256 lanes x 4 channels = 1024 channels.
// All geometry is block-uniform (SGPR/SALU, s_load_b128 descriptor fetch);
// taps are coalesced global_load_b128; output is streamed with non-temporal
// b128 stores so the write-once stream never evicts the L2-resident image.
// Fully unrolled px loop lets the compiler batch all 28 taps for max MLP —
// no explicit prefetch needed (the loads themselves are the prefetch).

#define POOL   7
#define N_ROIS 600
#define IMG_H  96
#define IMG_W  96
#define CH     1024

typedef float f32x4 __attribute__((ext_vector_type(4)));
typedef int   i32x4 __attribute__((ext_vector_type(4)));

__device__ __forceinline__ f32x4 lerp4(f32x4 a, f32x4 b, float f) {
    // Matches reference: a*(1-f) + b*f
    return a * (1.0f - f) + b * f;
}

__global__ __launch_bounds__(256) void roi_pool_bilinear_kernel(
        const float* __restrict__ img,   // (96, 96, 1024) NHWC, N=1
        const int*   __restrict__ rois,  // (600, 4) as (x, y, w, h)
        float*       __restrict__ out)   // (600, 7, 7, 1024)
{
    const int blk = blockIdx.x;          // 0 .. 600*7-1
    const int roi = blk / POOL;
    const int py  = blk - roi * POOL;

    // Uniform per-block ROI descriptor -> scalar load (s_load_b128).
    const i32x4 r  = *(const i32x4*)(rois + roi * 4);
    const int   rx = r.x;
    const int   ry = r.y;
    const int   rw = max(r.z, 1);
    const int   rh = max(r.w, 1);

    // ---- y taps (uniform per block; f32 division matches reference) ----
    const float in_y = (float)py * ((float)rh / (float)POOL);
    const int   y0   = (int)floorf(in_y);
    const float fy   = in_y - (float)y0;
    const int   y1   = min(y0 + 1, rh - 1);
    const int   row0 = min(max(ry + y0, 0), IMG_H - 1);
    const int   row1 = min(max(ry + y1, 0), IMG_H - 1);

    // ---- x taps for all 7 output columns (uniform, fully unrolled) ----
    const float sx = (float)rw / (float)POOL;
    int   c0[POOL], c1[POOL];
    float fx[POOL];
#pragma unroll
    for (int px = 0; px < POOL; ++px) {
        const float in_x = (float)px * sx;
        const int   x0   = (int)floorf(in_x);
        fx[px]           = in_x - (float)x0;
        const int   x1   = min(x0 + 1, rw - 1);
        c0[px] = min(max(rx + x0, 0), IMG_W - 1);
        c1[px] = min(max(rx + x1, 0), IMG_W - 1);
    }

    // ---- per-lane channel slice: lane owns 4 contiguous channels ----
    const int    lane  = threadIdx.x;               // 0..255
    const size_t coff  = (size_t)lane * 4;          // channel offset
    const float* base0 = img + (size_t)row0 * IMG_W * CH + coff;
    const float* base1 = img + (size_t)row1 * IMG_W * CH + coff;
    float*       orow  = out + ((size_t)roi * POOL + py) * POOL * CH + coff;

#pragma unroll
    for (int px = 0; px < POOL; ++px) {
        // Four coalesced b128 taps; block-uniform column offsets mean the
        // compiler can issue all 28 taps of the unrolled loop back-to-back.
        const f32x4 a = *(const f32x4*)(base0 + (size_t)c0[px] * CH);
        const f32x4 b = *(const f32x4*)(base0 + (size_t)c1[px] * CH);
        const f32x4 c = *(const f32x4*)(base1 + (size_t)c0[px] * CH);
        const f32x4 d = *(const f32x4*)(base1 + (size_t)c1[px] * CH);

        const f32x4 top = lerp4(a, b, fx[px]);
        const f32x4 bot = lerp4(c, d, fx[px]);
        const f32x4 v   = lerp4(top, bot, fy);

        // Write-once 120 MB stream: non-temporal so it doesn't evict the
        // L2-resident image that the gather reads depend on.
        __builtin_nontemporal_store(v, (f32x4*)(orow + (size_t)px * CH));
    }
}

extern "C" void kernel_launch(void* const* d_in, const int* in_sizes, int n_in,
                              void* d_out, int out_size, void* d_ws, size_t ws_size,
                              hipStream_t stream) {
    (void)in_sizes; (void)n_in; (void)out_size; (void)d_ws; (void)ws_size;
    const float* img  = (const float*)d_in[0];  // (1,96,96,1024) f32
    const int*   rois = (const int*)d_in[1];    // (1,600,4) i32
    float*       out  = (float*)d_out;          // (1,600,7,7,1024) f32

    dim3 grid(N_ROIS * POOL);   // 4200 blocks: one per (roi, output row)
    dim3 block(256);            // 8 wave32s; 4 channels per lane
    roi_pool_bilinear_kernel<<<grid, block, 0, stream>>>(img, rois, out);
}